// SparseHopfield_52570399703550
// MI455X (gfx1250) — compile-verified
//
#include <hip/hip_runtime.h>
#include <hip/hip_bf16.h>
#include <math.h>
#include <stdint.h>

typedef __attribute__((ext_vector_type(2))) float v2f;
typedef __attribute__((ext_vector_type(8))) float v8f;
typedef __attribute__((ext_vector_type(4))) unsigned int v4u;
typedef __attribute__((ext_vector_type(4))) int v4i;
typedef __attribute__((ext_vector_type(8))) int v8i;
typedef uint32_t u32;

#define RHO_C    1e-8f
#define EPS_C    1e-8f
#define THRESH_C 0.9f
#define MARKV    (-2)

// ---------------------------------------------------------------------
// Tensor Data Mover staging (CDNA5). Each wave issues a D# covering its
// share of a slab; TDM pad feature (+1 DWORD per 64 DWORDs) writes the
// 65-float padded LDS rows directly. Guarded: sync fallback otherwise.
// ---------------------------------------------------------------------
#if __has_builtin(__builtin_amdgcn_tensor_load_to_lds)
#define USE_TDM 1
#else
#define USE_TDM 0
#endif

#if USE_TDM
#define TDM_WAIT(n) __builtin_amdgcn_s_wait_tensorcnt((short)(n))
#else
#define TDM_WAIT(n) ((void)0)
#endif

__device__ __forceinline__ u32 lds_byte_off(const void* p) {
  // low 32 bits of a flat LDS address = byte offset within workgroup LDS
  return (u32)(uintptr_t)p;
}

// 2D tile load: tile_d0 elements/row (f32), tile_d1 rows (0 = 1D),
// row stride stride0 in elements. LDS gets +1 DWORD pad per 64 DWORDs.
__device__ __forceinline__ void tdm_issue(u32 lds_addr, const void* gptr,
                                          u32 tensor_d0, u32 tensor_d1,
                                          u32 tile_d0, u32 tile_d1,
                                          uint64_t stride0) {
#if USE_TDM
  uint64_t ga = (uint64_t)(uintptr_t)gptr;
  v4u g0;
  g0[0] = 1u;                                              // count=1, user D#
  g0[1] = lds_addr;                                        // lds_addr[31:0]
  g0[2] = (u32)ga;                                         // global_addr[31:0]
  g0[3] = ((u32)(ga >> 32) & 0x01FFFFFFu) | (2u << 30);    // addr[56:32]|type=2
  v8i g1;
  g1[0] = (int)((2u << 16)      // data_size = 4B
              | (1u << 20)      // pad_enable
              | (5u << 22)      // pad_interval: 64 DWORDs
              | (0u << 25));    // pad_amount: 1 DWORD
  g1[1] = (int)((tensor_d0 & 0xFFFFu) << 16);              // td0[15:0]
  g1[2] = (int)(((tensor_d0 >> 16) & 0xFFFFu) | ((tensor_d1 & 0xFFFFu) << 16));
  g1[3] = (int)(((tensor_d1 >> 16) & 0xFFFFu) | ((tile_d0 & 0xFFFFu) << 16));
  g1[4] = (int)(tile_d1 & 0xFFFFu);                        // tile_d1, tile_d2=0
  g1[5] = (int)(u32)(stride0 & 0xFFFFFFFFull);             // stride0[31:0]
  g1[6] = (int)(u32)((stride0 >> 32) & 0xFFFFull);         // stride0[47:32]
  g1[7] = 0;
  v4i z4 = {0, 0, 0, 0};
#if defined(__clang_major__) && __clang_major__ >= 23
  v8i z8 = {0, 0, 0, 0, 0, 0, 0, 0};
  __builtin_amdgcn_tensor_load_to_lds(g0, g1, z4, z4, z8, 0);
#else
  __builtin_amdgcn_tensor_load_to_lds(g0, g1, z4, z4, 0);
#endif
#else
  (void)lds_addr; (void)gptr; (void)tensor_d0; (void)tensor_d1;
  (void)tile_d0; (void)tile_d1; (void)stride0;
#endif
}

// =====================================================================
// Kernel 1: outer_forward
//   h0[b,f,m] = 0.5*sum_d (mm0[f,m,d]-0.5)*(xs[b,f,d]-0.5)
//              / (||m_f,m||*||x_b,f|| + rho) + 0.5
// One workgroup per field. 8 waves; each owns one 16x16 tile of the
// (32b x 64m) output via V_WMMA_F32_16X16X4_F32. Slabs staged by TDM.
// =====================================================================
__global__ __launch_bounds__(256) void outer_kernel(const float* __restrict__ xs,
                                                    const float* __restrict__ mm0,
                                                    float* __restrict__ h0) {
  const int f   = blockIdx.x;     // 0..4095
  const int tid = threadIdx.x;    // 0..255

  __shared__ float Ml[64 * 65];   // raw mm0[f], padded rows (65 f32)
  __shared__ float Xl[32 * 65];   // raw xs[:,f,:], padded rows
  __shared__ float mn[64];
  __shared__ float xn[32];

  const int wave = tid >> 5;
  const int lane = tid & 31;
  const float* mbase = mm0 + (size_t)f * 64 * 64;

#if USE_TDM
  // wave w moves 8 rows (512 f32) of the M slab + 4 rows of the X tile
  tdm_issue(lds_byte_off(&Ml[0]) + (u32)wave * 8 * 65 * 4,
            mbase + wave * 512,
            /*td0*/512, /*td1*/1, /*tile0*/512, /*tile1*/0, /*stride0*/512);
  tdm_issue(lds_byte_off(&Xl[0]) + (u32)wave * 4 * 65 * 4,
            xs + (size_t)(wave * 4) * 4096 * 64 + (size_t)f * 64,
            /*td0*/64, /*td1*/4, /*tile0*/64, /*tile1*/4,
            /*stride0*/(uint64_t)4096 * 64);
  TDM_WAIT(0);
#else
  #pragma unroll
  for (int i = 0; i < 16; ++i) {
    int idx = tid + i * 256;                 // 4096 elements
    Ml[(idx >> 6) * 65 + (idx & 63)] = mbase[idx];
  }
  #pragma unroll
  for (int i = 0; i < 8; ++i) {
    int idx = tid + i * 256;                 // 2048 elements
    int b = idx >> 6, d = idx & 63;
    Xl[b * 65 + d] = xs[(size_t)b * 4096 * 64 + (size_t)f * 64 + d];
  }
#endif
  __syncthreads();

  if (tid < 64) {
    float s = 0.f;
    #pragma unroll
    for (int d = 0; d < 64; ++d) { float v = Ml[tid * 65 + d] - 0.5f; s += v * v; }
    mn[tid] = sqrtf(s);
  }
  if (tid < 32) {
    float s = 0.f;
    #pragma unroll
    for (int d = 0; d < 64; ++d) { float v = Xl[tid * 65 + d] - 0.5f; s += v * v; }
    xn[tid] = sqrtf(s);
  }
  __syncthreads();

  const int lrow = lane & 15;           // A: M-row / B: N-col within tile
  const int kb   = (lane >> 4) * 2;     // K pair base (ISA 32-bit layout)
  const int b0   = (wave >> 2) * 16;    // batch-tile origin
  const int m0   = (wave & 3) * 16;     // mem-tile origin

  v8f acc = {};
  #pragma unroll
  for (int k0 = 0; k0 < 64; k0 += 4) {
    v2f a, b;
    a.x = Xl[(b0 + lrow) * 65 + k0 + kb]     - 0.5f;
    a.y = Xl[(b0 + lrow) * 65 + k0 + kb + 1] - 0.5f;
    b.x = Ml[(m0 + lrow) * 65 + k0 + kb]     - 0.5f;   // B = M^T
    b.y = Ml[(m0 + lrow) * 65 + k0 + kb + 1] - 0.5f;
    acc = __builtin_amdgcn_wmma_f32_16x16x4_f32(false, a, false, b,
                                                (short)0, acc, false, false);
  }

  const int mcol = m0 + lrow;
  const int hl   = lane >> 4;
  #pragma unroll
  for (int i = 0; i < 8; ++i) {
    int brow = b0 + i + 8 * hl;          // C/D row = vgpr + 8*(lane>=16)
    float v = acc[i] * 0.5f / (mn[mcol] * xn[brow] + RHO_C) + 0.5f;
    h0[(size_t)brow * 4096 * 64 + (size_t)f * 64 + mcol] = v;
  }
}

// =====================================================================
// Kernel 2: hidden_forward
//   x = maxi(child_x) (one nonzero per (b,c) row, stored as idx+val)
//   out[b,n,h] = sum_{c,k} mm[n,c,h,k]*x[b,n,c,k] / (8*||x_b,n|| + rho)
// One workgroup per node. mm[n,c] 16KB slabs stream through a
// double-buffered LDS staging area via TDM, overlapped with WMMA.
// =====================================================================
__global__ __launch_bounds__(256) void hidden_kernel(const float* __restrict__ mm,
                                                     const float* __restrict__ xin,
                                                     float* __restrict__ out,
                                                     int in_rows, int out_nodes) {
  const int n   = blockIdx.x;
  const int tid = threadIdx.x;
  const int wave = tid >> 5;
  const int lane = tid & 31;

  __shared__ float Ml[2][64 * 65];  // double-buffered mm[n,c] slab (h x k)
  __shared__ float mvS[32][8];      // max value per (b,c)
  __shared__ int   biS[32][8];      // argmax index per (b,c)
  __shared__ float xn[32];

  const float* slab0 = mm + (size_t)n * 8 * 64 * 64;

#if USE_TDM
  // Prefetch slab c=0 while we do the maxi scan (wave w: rows 8w..8w+7)
  tdm_issue(lds_byte_off(&Ml[0][0]) + (u32)wave * 8 * 65 * 4,
            slab0 + wave * 512, 512, 1, 512, 0, 512);
#endif

  {
    int b = tid >> 3, c = tid & 7;  // 256 threads = all (b,c)
    const float* row = xin + (size_t)b * in_rows * 64 + (size_t)(n * 8 + c) * 64;
    float best = row[0]; int bi = 0;
    for (int k = 1; k < 64; ++k) {
      float v = row[k];
      if (v > best) { best = v; bi = k; }   // first-occurrence argmax
    }
    mvS[b][c] = best;
    biS[b][c] = bi;
  }
  __syncthreads();
  if (tid < 32) {
    float s = 0.f;
    #pragma unroll
    for (int c = 0; c < 8; ++c) { float v = mvS[tid][c]; s += v * v; }
    xn[tid] = sqrtf(s);
  }

  const int lrow = lane & 15;
  const int kb   = (lane >> 4) * 2;
  const int b0   = (wave >> 2) * 16;
  const int h0   = (wave & 3) * 16;
  const int brA  = b0 + lrow;       // batch row this lane supplies for A

  v8f acc = {};
  for (int c = 0; c < 8; ++c) {
#if USE_TDM
    if (c < 7) {   // issue next slab into the other buffer
      tdm_issue(lds_byte_off(&Ml[(c + 1) & 1][0]) + (u32)wave * 8 * 65 * 4,
                slab0 + (size_t)(c + 1) * 64 * 64 + wave * 512,
                512, 1, 512, 0, 512);
      TDM_WAIT(1);                  // oldest (slab c) complete; c+1 in flight
    } else {
      TDM_WAIT(0);
    }
    __syncthreads();
    const float* MLc = Ml[c & 1];
#else
    {
      const float* mbase = slab0 + (size_t)c * 64 * 64;
      #pragma unroll
      for (int i = 0; i < 16; ++i) {
        int idx = tid + i * 256;
        Ml[0][(idx >> 6) * 65 + (idx & 63)] = mbase[idx];
      }
    }
    __syncthreads();
    const float* MLc = Ml[0];
#endif

    const int   bi = biS[brA][c];
    const float bv = mvS[brA][c];
    #pragma unroll
    for (int k0 = 0; k0 < 64; k0 += 4) {
      v2f a, b;
      a.x = (k0 + kb     == bi) ? bv : 0.f;   // maxi-sparse A row
      a.y = (k0 + kb + 1 == bi) ? bv : 0.f;
      b.x = MLc[(h0 + lrow) * 65 + k0 + kb];  // B[k][h] = mm[n,c,h,k]
      b.y = MLc[(h0 + lrow) * 65 + k0 + kb + 1];
      acc = __builtin_amdgcn_wmma_f32_16x16x4_f32(false, a, false, b,
                                                  (short)0, acc, false, false);
    }
    __syncthreads();   // all waves done reading before buffer reuse
  }

  const int hcol = h0 + lrow;
  const int hl   = lane >> 4;
  #pragma unroll
  for (int i = 0; i < 8; ++i) {
    int brow = b0 + i + 8 * hl;
    out[(size_t)brow * out_nodes * 64 + (size_t)n * 64 + hcol] =
        acc[i] / (8.f * xn[brow] + RHO_C);
  }
}

// =====================================================================
// Kernel 3: growth_argmaxi on h2 (32,64,64). One block per node (64),
// 64 threads. argmaxi, trigger, reserved-marking over a stable usage
// sort, MARK-to-back, expand (=first 32 cols), one-hot scatter.
// =====================================================================
__global__ __launch_bounds__(64) void growth_kernel(const float* __restrict__ h2,
                                                    const float* __restrict__ counts,
                                                    float* __restrict__ outp) {
  const int n   = blockIdx.x;     // 0..63
  const int tid = threadIdx.x;    // 0..63

  __shared__ float vmax[32];
  __shared__ int   amax[32];
  __shared__ int   trig[32];
  __shared__ int   reservedS[64];
  __shared__ int   sortidx[64];
  __shared__ int   availA[64];
  __shared__ int   movedA[64];
  __shared__ int   finalIdx[32];

  if (tid < 32) {
    const float* row = h2 + (size_t)tid * 64 * 64 + n * 64;
    float best = row[0]; int bi = 0;
    int cnt = (row[0] > THRESH_C) ? 1 : 0;
    for (int m = 1; m < 64; ++m) {
      float v = row[m];
      if (v > best) { best = v; bi = m; }
      cnt += (v > THRESH_C) ? 1 : 0;
    }
    vmax[tid] = best; amax[tid] = bi; trig[tid] = (cnt <= 0) ? 1 : 0;
  }
  __syncthreads();

  {   // reserved raw indices: argmax of a non-triggered batch row
    int r = 0;
    for (int b = 0; b < 32; ++b) r |= (!trig[b] && amax[b] == tid) ? 1 : 0;
    reservedS[tid] = r;
  }
  {   // stable ascending rank of counts[n, tid]
    float cv = counts[n * 64 + tid];
    int rank = 0;
    for (int j = 0; j < 64; ++j) {
      float cj = counts[n * 64 + j];
      rank += (cj < cv || (cj == cv && j < tid)) ? 1 : 0;
    }
    sortidx[rank] = tid;    // ranks unique -> conflict-free scatter
  }
  __syncthreads();

  availA[tid] = reservedS[sortidx[tid]] ? MARKV : sortidx[tid];
  __syncthreads();

  {   // stable move of MARK entries to the back
    int a = availA[tid];
    int pos;
    if (a != MARKV) {
      pos = 0;
      for (int q = 0; q < tid; ++q) pos += (availA[q] != MARKV) ? 1 : 0;
    } else {
      int nm = 0, before = 0;
      for (int q = 0; q < 64; ++q) nm += (availA[q] != MARKV) ? 1 : 0;
      for (int q = 0; q < tid; ++q) before += (availA[q] == MARKV) ? 1 : 0;
      pos = nm + before;
    }
    movedA[pos] = a;
  }
  __syncthreads();

  if (tid < 32) {   // expand_for_batches(b=32<=64) == first 32 columns
    int av = movedA[tid];
    finalIdx[tid] = (av == MARKV) ? sortidx[tid] : av;
  }
  __syncthreads();

  for (int b = 0; b < 32; ++b) {
    float val;
    if (trig[b]) {
      int idx = finalIdx[b];
      float v = h2[(size_t)b * 64 * 64 + n * 64 + idx];
      if (v == 0.f) v = 1.f;
      float av = fabsf(v);
      val = (tid == idx) ? (av / (av - EPS_C)) : 0.f;
    } else {
      float av = fabsf(vmax[b]);
      val = (tid == amax[b]) ? (av / (av - EPS_C)) : 0.f;
    }
    outp[(size_t)b * 64 * 64 + n * 64 + tid] = val;
  }
}

// =====================================================================
extern "C" void kernel_launch(void* const* d_in, const int* in_sizes, int n_in,
                              void* d_out, int out_size, void* d_ws, size_t ws_size,
                              hipStream_t stream) {
  (void)in_sizes; (void)n_in; (void)out_size; (void)ws_size;

  const float* xs     = (const float*)d_in[0];   // (32,4096,64)
  const float* mm0    = (const float*)d_in[1];   // (4096,64,64)
  const float* mm1    = (const float*)d_in[2];   // (512,8,64,64)
  const float* mm2    = (const float*)d_in[3];   // (64,8,64,64)
  const float* counts = (const float*)d_in[4];   // (64,64)
  float* out = (float*)d_out;                    // (32,64,64)

  float* h0 = (float*)d_ws;                      // 32*4096*64 floats (32 MB)
  float* h1 = h0 + (size_t)32 * 4096 * 64;       // 32*512*64 floats (4 MB)
  float* h2 = h1 + (size_t)32 * 512 * 64;        // 32*64*64 floats (0.5 MB)

  outer_kernel <<<4096, 256, 0, stream>>>(xs, mm0, h0);
  hidden_kernel<<< 512, 256, 0, stream>>>(mm1, h0, h1, 4096, 512);
  hidden_kernel<<<  64, 256, 0, stream>>>(mm2, h1, h2,  512,  64);
  growth_kernel<<<  64,  64, 0, stream>>>(h2, counts, out);
}